// GCN_27633819583013
// MI455X (gfx1250) — compile-verified
//
#include <hip/hip_runtime.h>

typedef __attribute__((ext_vector_type(2))) float v2f;
typedef __attribute__((ext_vector_type(8))) float v8f;

#define D_INF 128
#define D_HID 128
#define D_OUTF 64
#define NGRAPH 100

// ---------------- degree + normalization ----------------
__global__ void deg_kernel(const int* __restrict__ src, const int* __restrict__ dst,
                           float* __restrict__ outdeg, float* __restrict__ indeg, int E) {
    int e = blockIdx.x * blockDim.x + threadIdx.x;
    if (e < E) {
        atomicAdd(&outdeg[src[e]], 1.0f);
        atomicAdd(&indeg[dst[e]], 1.0f);
    }
}

__global__ void dinv_kernel(float* __restrict__ a, float* __restrict__ b, int N) {
    int i = blockIdx.x * blockDim.x + threadIdx.x;
    if (i < N) {
        a[i] = rsqrtf(fmaxf(a[i], 1.0f));
        b[i] = rsqrtf(fmaxf(b[i], 1.0f));
    }
}

// ---------------- WMMA f32 GEMM: H = (X * scale_row) @ W ----------------
// One wave32 computes one 16x16 tile of H. K stepped by 4 with
// v_wmma_f32_16x16x4_f32. A layout (ISA 7.12.2, 32-bit A 16x4):
//   lanes 0-15: M=lane, VGPR0=K0, VGPR1=K1 ; lanes 16-31: M=lane-16, K+=2.
__global__ void gemm_wmma_kernel(const float* __restrict__ X,
                                 const float* __restrict__ rowscale,
                                 const float* __restrict__ W,
                                 float* __restrict__ H,
                                 int N, int K, int Dn) {
    const int lane = threadIdx.x & 31;
    const int wave = threadIdx.x >> 5;
    const int tilesM = (N + 15) >> 4;
    const int tilesN = Dn >> 4;
    const int tile = blockIdx.x * (blockDim.x >> 5) + wave;
    if (tile >= tilesM * tilesN) return;            // wave-uniform: EXEC all-1 for WMMA
    const int tm = tile / tilesN;
    const int tn = tile % tilesN;

    const int row  = tm * 16 + (lane & 15);
    const int rowc = row < N ? row : (N - 1);       // clamp tail reads
    const int col  = tn * 16 + (lane & 15);
    const int khi  = (lane >> 4) << 1;              // 0 for lanes 0-15, 2 for 16-31

    const float* __restrict__ xr = X + (size_t)rowc * K;
    const float s = rowscale[rowc];

    v8f c = {};
#pragma unroll 4
    for (int k = 0; k < K; k += 4) {
        const int kb = k + khi;
        v2f a, b;
        a.x = xr[kb + 0] * s;
        a.y = xr[kb + 1] * s;
        b.x = W[(size_t)(kb + 0) * Dn + col];
        b.y = W[(size_t)(kb + 1) * Dn + col];
        c = __builtin_amdgcn_wmma_f32_16x16x4_f32(
                /*neg_a=*/false, a, /*neg_b=*/false, b,
                /*c_mod=*/(short)0, c, /*reuse_a=*/false, /*reuse_b=*/false);
    }

    // C/D layout: VGPR v -> M = tm*16 + v + (lane>=16 ? 8 : 0), N = lane&15
    const int rbase = tm * 16 + ((lane >> 4) << 3);
#pragma unroll
    for (int v = 0; v < 8; ++v) {
        const int r = rbase + v;
        if (r < N) H[(size_t)r * Dn + col] = c[v];
    }
}

// ---------------- SpMM: AGG[dst] += H[src] (wave per edge, lane-strided) ----
__global__ void spmm_kernel(const float* __restrict__ H,
                            const int* __restrict__ src, const int* __restrict__ dst,
                            float* __restrict__ AGG, int E, int D) {
    const int wavesPerBlock = blockDim.x >> 5;
    const int lane = threadIdx.x & 31;
    int w = blockIdx.x * wavesPerBlock + (threadIdx.x >> 5);
    const int nw = gridDim.x * wavesPerBlock;
    for (int e = w; e < E; e += nw) {
        const int sN = src[e];
        const int dN = dst[e];
        const float* __restrict__ hs = H + (size_t)sN * D;
        float* __restrict__ ad = AGG + (size_t)dN * D;
        for (int j = lane; j < D; j += 32)
            atomicAdd(&ad[j], hs[j]);
    }
}

// ---------------- post: Y = AGG * dinv_dst + b (+ LeakyReLU 0.1) -----------
__global__ void post_kernel(const float* __restrict__ AGG,
                            const float* __restrict__ dinv_dst,
                            const float* __restrict__ bias,
                            float* __restrict__ Y, int N, int D, int lrelu) {
    int idx = blockIdx.x * blockDim.x + threadIdx.x;
    if (idx >= N * D) return;
    const int i = idx / D;
    const int j = idx - i * D;
    float v = AGG[idx] * dinv_dst[i] + bias[j];
    if (lrelu) v = v >= 0.0f ? v : 0.1f * v;
    Y[idx] = v;
}

// ---------------- readout: per-node dot + segment sum ----------------------
__global__ void readout_kernel(const float* __restrict__ F,
                               const float* __restrict__ Wout,
                               const float* __restrict__ bout,
                               const int* __restrict__ gid,
                               float* __restrict__ gsum, float* __restrict__ gcnt, int N) {
    int i = blockIdx.x * blockDim.x + threadIdx.x;
    if (i >= N) return;
    const float* __restrict__ fr = F + (size_t)i * D_OUTF;
    float acc = 0.0f;
#pragma unroll 8
    for (int j = 0; j < D_OUTF; ++j) acc += fr[j] * Wout[j];
    acc += bout[0];
    const int g = gid[i];
    atomicAdd(&gsum[g], acc);
    atomicAdd(&gcnt[g], 1.0f);
}

__global__ void mean_kernel(const float* __restrict__ gsum,
                            const float* __restrict__ gcnt,
                            float* __restrict__ out) {
    int g = blockIdx.x * blockDim.x + threadIdx.x;
    if (g < NGRAPH) out[g] = gsum[g] / fmaxf(gcnt[g], 1.0f);
}

// ---------------------------------------------------------------------------
extern "C" void kernel_launch(void* const* d_in, const int* in_sizes, int n_in,
                              void* d_out, int out_size, void* d_ws, size_t ws_size,
                              hipStream_t stream) {
    const float* in_feat = (const float*)d_in[0];
    const float* W1   = (const float*)d_in[1];
    const float* b1   = (const float*)d_in[2];
    const float* W2   = (const float*)d_in[3];
    const float* b2   = (const float*)d_in[4];
    const float* W3   = (const float*)d_in[5];
    const float* b3   = (const float*)d_in[6];
    const float* Wout = (const float*)d_in[7];
    const float* bout = (const float*)d_in[8];
    const int*   src  = (const int*)d_in[9];
    const int*   dst  = (const int*)d_in[10];
    const int*   gid  = (const int*)d_in[11];
    const int N = in_sizes[11];        // graph_ids has N entries
    const int E = in_sizes[9];

    float* ws    = (float*)d_ws;
    float* bufA  = ws;                               // N * 128
    float* bufB  = bufA + (size_t)N * D_HID;         // N * 128
    float* dinvs = bufB + (size_t)N * D_HID;         // N
    float* dinvd = dinvs + N;                        // N
    float* gsum  = dinvd + N;                        // NGRAPH
    float* gcnt  = gsum + NGRAPH;                    // NGRAPH

    // ---- degrees -> dinv ----
    hipMemsetAsync(dinvs, 0, sizeof(float) * (size_t)N * 2, stream);
    deg_kernel<<<(E + 255) / 256, 256, 0, stream>>>(src, dst, dinvs, dinvd, E);
    dinv_kernel<<<(N + 255) / 256, 256, 0, stream>>>(dinvs, dinvd, N);

    const int tilesM = (N + 15) / 16;
    const int spmmBlocks = 2048;                     // 8 waves/block, grid-stride over E
    const int postBlocks128 = (N * D_HID + 255) / 256;
    const int postBlocks64  = (N * D_OUTF + 255) / 256;

    auto gconv = [&](const float* F, const float* W, const float* b,
                     float* Hbuf, float* Abuf, float* Fnext, int Dn, int lrelu) {
        const int tiles = tilesM * (Dn / 16);
        gemm_wmma_kernel<<<(tiles + 7) / 8, 256, 0, stream>>>(F, dinvs, W, Hbuf, N, D_HID, Dn);
        hipMemsetAsync(Abuf, 0, sizeof(float) * (size_t)N * Dn, stream);
        spmm_kernel<<<spmmBlocks, 256, 0, stream>>>(Hbuf, src, dst, Abuf, E, Dn);
        post_kernel<<<(Dn == D_HID ? postBlocks128 : postBlocks64), 256, 0, stream>>>(
            Abuf, dinvd, b, Fnext, N, Dn, lrelu);
    };

    // Layer 1: F=in_feat, H=bufA, AGG=bufB, Fnext=bufA
    gconv(in_feat, W1, b1, bufA, bufB, bufA, D_HID, 1);
    // Layer 2: F=bufA,   H=bufB, AGG=bufA, Fnext=bufB
    gconv(bufA,    W2, b2, bufB, bufA, bufB, D_HID, 1);
    // Layer 3 (conv2 applied twice): F=bufB, H=bufA, AGG=bufB, Fnext=bufA
    gconv(bufB,    W2, b2, bufA, bufB, bufA, D_HID, 1);
    // Layer 4: F=bufA, H=bufB, AGG=bufA, Fnext=bufB, Dout=64, no lrelu
    gconv(bufA,    W3, b3, bufB, bufA, bufB, D_OUTF, 0);

    // ---- readout: per-node scalar, segment mean ----
    hipMemsetAsync(gsum, 0, sizeof(float) * NGRAPH * 2, stream);
    readout_kernel<<<(N + 255) / 256, 256, 0, stream>>>(bufB, Wout, bout, gid, gsum, gcnt, N);
    mean_kernel<<<1, 128, 0, stream>>>(gsum, gcnt, (float*)d_out);
}